// DecisionSufficientAbstraction_21766894256562
// MI455X (gfx1250) — compile-verified
//
#include <hip/hip_runtime.h>

typedef __attribute__((ext_vector_type(2))) float v2f;
typedef __attribute__((ext_vector_type(4))) float f4;
typedef __attribute__((ext_vector_type(8))) float v8f;

#define B_ 128
#define N_ 2048
#define D_ 256
#define K_ 64
#define RSQRT_D 0.0625f   // 1/sqrt(256)

// ---------------------------------------------------------------------------
// Workspace layout (in floats)
// ---------------------------------------------------------------------------
#define WS_QUERY   0                         // B*D
#define WS_V       (WS_QUERY + B_*D_)        // B*D
#define WS_C       (WS_V + B_*D_)            // B
#define WS_SCORES  (WS_C + B_)               // B*N
#define WS_GSUM    (WS_SCORES + B_*N_)       // B*D  (atomic accum, must be zeroed)
#define WS_GCNT    (WS_GSUM + B_*D_)         // B    (atomic accum, must be zeroed)
#define WS_SELIDX  (WS_GCNT + B_)            // B*K  (int)

// ---------------------------------------------------------------------------
// Kernel 1: query[b,e] = sum_d ego[b,d]*Wq[e,d] + bq[e]
// WMMA f32 16x16x4. A = ego (M=b, K=d, row stride N*D), B[k=d, n=e] = Wq[e,d].
// One wave per 16x16 output tile. Grid (B/16, D/16), block 32.
// Layout assumption (per ISA 7.12.2, 32-bit A 16x4): lanes 0-15 hold K={0,1}
// in VGPR0/1, lanes 16-31 hold K={2,3}; B mirrors with lane%16 = column.
// ---------------------------------------------------------------------------
__global__ void qproj_wmma(const float* __restrict__ latent,
                           const float* __restrict__ Wq,
                           const float* __restrict__ bq,
                           float* __restrict__ query) {
  const int Mb   = blockIdx.x * 16;
  const int Nb   = blockIdx.y * 16;
  const int lane = threadIdx.x & 31;
  const int half = lane >> 4;
  const int idx  = lane & 15;
  const long arow = (long)(Mb + idx) * (long)(N_ * D_);  // ego row (n=0)
  const long brow = (long)(Nb + idx) * D_;               // Wq row e
  v8f c = {};
  for (int k = 0; k < D_; k += 4) {
    const int kk = k + 2 * half;
    v2f a = *(const v2f*)(latent + arow + kk);  // A[M=Mb+idx, K=kk,kk+1]
    v2f b = *(const v2f*)(Wq + brow + kk);      // B[K=kk,kk+1, N=Nb+idx] = Wq[e, d]
    c = __builtin_amdgcn_wmma_f32_16x16x4_f32(false, a, false, b,
                                              (short)0, c, false, false);
  }
#pragma unroll
  for (int r = 0; r < 8; ++r) {
    const int row = Mb + r + 8 * half;
    const int col = Nb + idx;
    query[row * D_ + col] = c[r] + bq[col];
  }
}

// ---------------------------------------------------------------------------
// Kernel 2: v[b,d] = (sum_e query[b,e]*Wk[e,d]) / sqrt(D) + Ws[d]
// A = query (M=b, K=e), B[k=e, n=d] = Wk[e,d] (natural row-major, stride D).
// ---------------------------------------------------------------------------
__global__ void vproj_wmma(const float* __restrict__ query,
                           const float* __restrict__ Wk,
                           const float* __restrict__ Ws,
                           float* __restrict__ v) {
  const int Mb   = blockIdx.x * 16;
  const int Nb   = blockIdx.y * 16;
  const int lane = threadIdx.x & 31;
  const int half = lane >> 4;
  const int idx  = lane & 15;
  const long arow = (long)(Mb + idx) * D_;
  v8f c = {};
  for (int k = 0; k < D_; k += 4) {
    const int kk = k + 2 * half;
    v2f a = *(const v2f*)(query + arow + kk);
    v2f b;
    b.x = Wk[(long)(kk + 0) * D_ + Nb + idx];
    b.y = Wk[(long)(kk + 1) * D_ + Nb + idx];
    c = __builtin_amdgcn_wmma_f32_16x16x4_f32(false, a, false, b,
                                              (short)0, c, false, false);
  }
#pragma unroll
  for (int r = 0; r < 8; ++r) {
    const int row = Mb + r + 8 * half;
    const int col = Nb + idx;
    v[row * D_ + col] = c[r] * RSQRT_D + Ws[col];
  }
}

// ---------------------------------------------------------------------------
// Kernel 3: c[b] = (query[b,:] . bk) / sqrt(D) + bs[0].  One wave per batch.
// ---------------------------------------------------------------------------
__global__ void cvec_kernel(const float* __restrict__ query,
                            const float* __restrict__ bk,
                            const float* __restrict__ bs,
                            float* __restrict__ cout) {
  const int b = blockIdx.x;
  const int lane = threadIdx.x;
  float s = 0.0f;
#pragma unroll
  for (int j = 0; j < 8; ++j) {
    const int e = lane * 8 + j;
    s += query[b * D_ + e] * bk[e];
  }
#pragma unroll
  for (int m = 16; m >= 1; m >>= 1) s += __shfl_xor(s, m, 32);
  if (lane == 0) cout[b] = s * RSQRT_D + bs[0];
}

// ---------------------------------------------------------------------------
// Kernel 4 (the 256 MB streaming pass):
//   scores[b,n] = mask ? latent[b,n,:].v[b] + c[b] : -inf
//   gsum[b,:]  += mask * latent[b,n,:]  (f32 global atomics, wave-accumulated)
//   gcnt[b]    += mask
// Grid: B*8 blocks (8 n-chunks of 256 rows), 256 threads = 8 waves,
// wave-per-row, lane owns 8 contiguous d's (two b128 loads per row).
// ---------------------------------------------------------------------------
__global__ void scores_gsum_kernel(const float* __restrict__ latent,
                                   const int* __restrict__ mask,
                                   const float* __restrict__ v,
                                   const float* __restrict__ cvec,
                                   float* __restrict__ scores,
                                   float* __restrict__ gsum,
                                   float* __restrict__ gcnt) {
  const int b     = blockIdx.x >> 3;
  const int chunk = blockIdx.x & 7;
  const int wave  = threadIdx.x >> 5;
  const int lane  = threadIdx.x & 31;
  const long base = (long)b * N_ * D_;

  float vreg[8];
#pragma unroll
  for (int j = 0; j < 8; ++j) vreg[j] = v[b * D_ + lane * 8 + j];
  const float cb = cvec[b];

  float acc[8] = {0, 0, 0, 0, 0, 0, 0, 0};
  int cnt = 0;
  const int n0 = chunk * 256 + wave * 32;

  for (int r = 0; r < 32; ++r) {
    const int n = n0 + r;
    const float* row = latent + base + (long)n * D_ + lane * 8;
    f4 x0 = *(const f4*)(row);
    f4 x1 = *(const f4*)(row + 4);
    const int m = mask[b * N_ + n];
    float p = x0.x * vreg[0] + x0.y * vreg[1] + x0.z * vreg[2] + x0.w * vreg[3]
            + x1.x * vreg[4] + x1.y * vreg[5] + x1.z * vreg[6] + x1.w * vreg[7];
    if (m) {
      acc[0] += x0.x; acc[1] += x0.y; acc[2] += x0.z; acc[3] += x0.w;
      acc[4] += x1.x; acc[5] += x1.y; acc[6] += x1.z; acc[7] += x1.w;
      ++cnt;
    }
#pragma unroll
    for (int sft = 16; sft >= 1; sft >>= 1) p += __shfl_xor(p, sft, 32);
    if (lane == 0) scores[b * N_ + n] = m ? (p + cb) : -__builtin_inff();
  }
#pragma unroll
  for (int j = 0; j < 8; ++j)
    unsafeAtomicAdd(&gsum[b * D_ + lane * 8 + j], acc[j]);
  if (lane == 0) unsafeAtomicAdd(&gcnt[b], (float)cnt);
}

// ---------------------------------------------------------------------------
// Kernel 5: per-batch top-64 (jax tie-break: smallest index), selected mask,
// indices, and masked softmax importance. One block (256 thr) per batch.
// ---------------------------------------------------------------------------
__global__ void topk_kernel(const float* __restrict__ scores,
                            const int* __restrict__ mask,
                            float* __restrict__ out_mask,
                            float* __restrict__ out_idx,
                            float* __restrict__ out_imp,
                            int* __restrict__ selIdxWs) {
  __shared__ float s[N_];
  __shared__ float rv[256];
  __shared__ int   ri[256];
  __shared__ float selS[K_];
  __shared__ int   selI[K_];
  __shared__ float imp[K_];
  __shared__ float stats[2];

  const int b = blockIdx.x;
  const int tid = threadIdx.x;

  for (int i = tid; i < N_; i += 256) s[i] = scores[b * N_ + i];
  __syncthreads();

  for (int k = 0; k < K_; ++k) {
    float best = s[tid];
    int bi = tid;
    for (int i = tid + 256; i < N_; i += 256) {
      const float vv = s[i];
      if (vv > best) { best = vv; bi = i; }   // strict > keeps smallest index
    }
    rv[tid] = best; ri[tid] = bi;
    __syncthreads();
    for (int off = 128; off >= 1; off >>= 1) {
      if (tid < off) {
        const float v2 = rv[tid + off];
        const int i2 = ri[tid + off];
        if (v2 > rv[tid] || (v2 == rv[tid] && i2 < ri[tid])) {
          rv[tid] = v2; ri[tid] = i2;
        }
      }
      __syncthreads();
    }
    if (tid == 0) {
      selI[k] = ri[0];
      selS[k] = rv[0];
      s[ri[0]] = -__builtin_inff();
    }
    __syncthreads();
  }

  if (tid < K_) {
    const int ii = selI[tid];
    const int m = mask[b * N_ + ii];
    imp[tid] = m ? selS[tid] : -1.0e9f;
    out_mask[b * K_ + tid] = m ? 1.0f : 0.0f;
    out_idx[b * K_ + tid] = (float)ii;
    selIdxWs[b * K_ + tid] = ii;
  }
  __syncthreads();
  if (tid == 0) {
    float mx = imp[0];
    for (int k = 1; k < K_; ++k) mx = fmaxf(mx, imp[k]);
    float sum = 0.0f;
    for (int k = 0; k < K_; ++k) sum += __expf(imp[k] - mx);
    stats[0] = mx; stats[1] = sum;
  }
  __syncthreads();
  if (tid < K_)
    out_imp[b * K_ + tid] = __expf(imp[tid] - stats[0]) / stats[1];
}

// ---------------------------------------------------------------------------
// Kernel 6: gather selected tokens. One wave per (b,k) row, lane copies 8 f32.
// ---------------------------------------------------------------------------
__global__ void gather_kernel(const float* __restrict__ latent,
                              const int* __restrict__ selIdx,
                              float* __restrict__ out_tok) {
  const int row  = blockIdx.x * 8 + (threadIdx.x >> 5);  // b*K + k
  const int lane = threadIdx.x & 31;
  const int b = row / K_;
  const int idx = selIdx[row];
  const float* src = latent + ((long)b * N_ + idx) * D_ + lane * 8;
  float* dst = out_tok + (long)row * D_ + lane * 8;
  *(f4*)(dst)     = *(const f4*)(src);
  *(f4*)(dst + 4) = *(const f4*)(src + 4);
}

// ---------------------------------------------------------------------------
// Kernel 7: global_latent[b,d] = gsum[b,d] / gcnt[b]
// ---------------------------------------------------------------------------
__global__ void finalize_gl(const float* __restrict__ gsum,
                            const float* __restrict__ gcnt,
                            float* __restrict__ out_gl) {
  const int i = blockIdx.x * 256 + threadIdx.x;
  out_gl[i] = gsum[i] / gcnt[i >> 8];  // D_ == 256
}

// ---------------------------------------------------------------------------
extern "C" void kernel_launch(void* const* d_in, const int* in_sizes, int n_in,
                              void* d_out, int out_size, void* d_ws, size_t ws_size,
                              hipStream_t stream) {
  const float* latent = (const float*)d_in[0];
  const int*   tmask  = (const int*)d_in[1];
  const float* Wq     = (const float*)d_in[2];
  const float* bq     = (const float*)d_in[3];
  const float* Wk     = (const float*)d_in[4];
  const float* bk     = (const float*)d_in[5];
  const float* Ws     = (const float*)d_in[6];
  const float* bs     = (const float*)d_in[7];

  float* ws = (float*)d_ws;
  float* q_ws     = ws + WS_QUERY;
  float* v_ws     = ws + WS_V;
  float* c_ws     = ws + WS_C;
  float* sc_ws    = ws + WS_SCORES;
  float* gsum_ws  = ws + WS_GSUM;
  float* gcnt_ws  = ws + WS_GCNT;
  int*   sel_ws   = (int*)(ws + WS_SELIDX);

  // Output layout (flat, in reference return order, all as float)
  float* out_tok  = (float*)d_out;                       // (B,K,D)
  float* out_mask = out_tok + (size_t)B_ * K_ * D_;      // (B,K)
  float* out_idx  = out_mask + (size_t)B_ * K_;          // (B,K)
  float* out_imp  = out_idx + (size_t)B_ * K_;           // (B,K)
  float* out_gl   = out_imp + (size_t)B_ * K_;           // (B,D)

  // Zero the atomic accumulators (ws is poisoned, and must be re-zeroed
  // every call for determinism). Memset nodes are graph-capturable.
  hipMemsetAsync(gsum_ws, 0, (size_t)(B_ * D_ + B_) * sizeof(float), stream);

  // 1) query = ego @ Wq^T + bq           (WMMA f32 16x16x4)
  qproj_wmma<<<dim3(B_ / 16, D_ / 16), 32, 0, stream>>>(latent, Wq, bq, q_ws);
  // 2) v = (query @ Wk)/sqrt(D) + Ws     (WMMA f32 16x16x4)
  vproj_wmma<<<dim3(B_ / 16, D_ / 16), 32, 0, stream>>>(q_ws, Wk, Ws, v_ws);
  // 3) c[b] = (query.bk)/sqrt(D) + bs
  cvec_kernel<<<B_, 32, 0, stream>>>(q_ws, bk, bs, c_ws);
  // 4) scores + masked sums (single 256 MB streaming pass over latent)
  scores_gsum_kernel<<<B_ * 8, 256, 0, stream>>>(latent, tmask, v_ws, c_ws,
                                                 sc_ws, gsum_ws, gcnt_ws);
  // 5) top-K, selected mask/indices, importance softmax
  topk_kernel<<<B_, 256, 0, stream>>>(sc_ws, tmask, out_mask, out_idx,
                                      out_imp, sel_ws);
  // 6) gather selected tokens
  gather_kernel<<<(B_ * K_) / 8, 256, 0, stream>>>(latent, sel_ws, out_tok);
  // 7) global_latent = masked mean
  finalize_gl<<<(B_ * D_) / 256, 256, 0, stream>>>(gsum_ws, gcnt_ws, out_gl);
}